// PerceptualMelLoss_84593675862632
// MI455X (gfx1250) — compile-verified
//
#include <hip/hip_runtime.h>
#include <stdint.h>

#define N_MELS   80
#define BATCH    64
#define TLEN     4000
#define TT       64          // frames per tile
#define SLOTS    (TT + 2)    // +2 halo rows for delta/delta2
#define NTHREADS 96          // 3 waves, lane = mel bin

typedef float v2f __attribute__((ext_vector_type(2)));
typedef float v8f __attribute__((ext_vector_type(8)));
typedef unsigned int v4u __attribute__((ext_vector_type(4)));
typedef int v4i __attribute__((ext_vector_type(4)));
typedef int v8i __attribute__((ext_vector_type(8)));

// Exact f32 sum of one value across the 32 lanes of a wave using
// V_WMMA_F32_16X16X4_F32: A[m][0]=x(lane m), A[m][2]=x(lane m+16), B=ones
// => D[m][n] = x(m) + x(m+16). Sum the 8 D VGPRs, fold halves with one shuffle.
__device__ __forceinline__ float wave_sum32(float x) {
  v2f a; a[0] = x;    a[1] = 0.0f;
  v2f b; b[0] = 1.0f; b[1] = 1.0f;
  v8f c = {};
  v8f d = __builtin_amdgcn_wmma_f32_16x16x4_f32(false, a, false, b,
                                                (short)0, c, false, false);
  float s = (d[0] + d[1]) + (d[2] + d[3]) + (d[4] + d[5]) + (d[6] + d[7]);
  s += __shfl_xor(s, 16, 32);
  return s;
}

// 2-D TDM load: nrows rows of N_MELS f32 from gsrc_row into LDS at lds_off.
__device__ __forceinline__ void tdm_load_2d(uint32_t lds_off, const float* gsrc_row,
                                            int nrows) {
  uint64_t ga = (uint64_t)(uintptr_t)gsrc_row;
  const uint32_t td0 = N_MELS;           // tensor_dim0 (elements)
  const uint32_t td1 = BATCH * TLEN;     // tensor_dim1 (rows)
  v4u g0;
  g0[0] = 1u;                                        // count=1, user mode
  g0[1] = lds_off;                                   // lds_addr (bytes)
  g0[2] = (uint32_t)ga;                              // global_addr[31:0]
  g0[3] = (uint32_t)(ga >> 32) | 0x80000000u;        // global_addr[56:32] | type=2
  v8i g1;
  g1[0] = (int)(2u << 16);                                           // data_size=4B
  g1[1] = (int)((td0 & 0xffffu) << 16);                              // tensor_dim0 lo
  g1[2] = (int)(((td0 >> 16) & 0xffffu) | ((td1 & 0xffffu) << 16));  // dim0 hi | dim1 lo
  g1[3] = (int)(((td1 >> 16) & 0xffffu) | ((uint32_t)N_MELS << 16)); // dim1 hi | tile_dim0
  g1[4] = (int)((uint32_t)nrows & 0xffffu);                          // tile_dim1
  g1[5] = (int)N_MELS;                                               // tensor_dim0_stride
  g1[6] = 0;
  g1[7] = 0;
  v4i z4 = {0, 0, 0, 0};
  v8i z8 = {0, 0, 0, 0, 0, 0, 0, 0};
  __builtin_amdgcn_tensor_load_to_lds(g0, g1, z4, z4, z8, 0);
}

__global__ void mel_loss_init(float* acc) {
  if (threadIdx.x < 16) acc[threadIdx.x] = 0.0f;
}

__global__ __launch_bounds__(NTHREADS)
void mel_loss_main(const float* __restrict__ pred, const float* __restrict__ tgt,
                   const float* __restrict__ mask, const float* __restrict__ bw,
                   float* __restrict__ acc) {
  __shared__ float shP[SLOTS * N_MELS];
  __shared__ float shT[SLOTS * N_MELS];
  __shared__ float ePart[TT * 4];
  __shared__ float red[3 * 8];

  const int b   = blockIdx.x;
  const int t0  = blockIdx.y * TT;
  const int tid = (int)threadIdx.x;
  const int d   = tid;
  const bool activeD = d < N_MELS;

  // tile geometry (uniform): slots 0,1 = halo rows t0-2, t0-1 when t0>0
  int r0, slot0, nrows;
  if (t0 == 0) {
    r0 = b * TLEN; slot0 = 2; nrows = TT;
  } else {
    r0 = b * TLEN + t0 - 2;
    slot0 = 0;
    int rem = (b * TLEN + TLEN) - r0;          // rows left in this batch
    nrows = rem < SLOTS ? rem : SLOTS;
  }

  if (tid < 32) {  // one wave issues the DMAs (TDM ignores EXEC; avoid dup issue)
    uint32_t ldsP = (uint32_t)(uintptr_t)shP + (uint32_t)slot0 * N_MELS * 4u;
    uint32_t ldsT = (uint32_t)(uintptr_t)shT + (uint32_t)slot0 * N_MELS * 4u;
    tdm_load_2d(ldsP, pred + (size_t)r0 * N_MELS, nrows);
    tdm_load_2d(ldsT, tgt  + (size_t)r0 * N_MELS, nrows);
    __builtin_amdgcn_s_wait_tensorcnt(0);
  }
  __syncthreads();

  const float w_d = activeD ? bw[d] : 0.0f;
  const size_t mbase = (size_t)b * TLEN;

  float a_l1 = 0.f, a_band = 0.f, a_diff2 = 0.f, a_tgt2 = 0.f;
  float a_delta = 0.f, a_delta2 = 0.f;
  float c_m = 0.f, c_dm = 0.f, c_d2m = 0.f;
  float pp1 = 0.f, pp2 = 0.f, tp1 = 0.f, tp2 = 0.f;
  float m1 = 0.f, m2 = 0.f;

  if (t0 > 0) {  // warm the register carry from halo rows
    if (activeD) {
      pp2 = shP[0 * N_MELS + d]; tp2 = shT[0 * N_MELS + d];
      pp1 = shP[1 * N_MELS + d]; tp1 = shT[1 * N_MELS + d];
    }
    m2 = mask[mbase + t0 - 2];
    m1 = mask[mbase + t0 - 1];
  }

  for (int i = 0; i < TT; ++i) {
    const int t = t0 + i;
    if (t >= TLEN) break;                      // uniform
    const int slot = i + 2;
    const float p = activeD ? shP[slot * N_MELS + d] : 0.0f;
    const float g = activeD ? shT[slot * N_MELS + d] : 0.0f;
    const float m = mask[mbase + t];

    const float diff = p - g;
    const float ad = fabsf(diff);
    a_l1   += ad * m;
    a_band += ad * w_d * m;
    const float dm = diff * m;
    a_diff2 += dm * dm;
    const float gm = g * m;
    a_tgt2  += gm * gm;

    if (t >= 1) {
      const float d1 = (p - pp1) - (g - tp1);
      a_delta += fabsf(d1) * (m * m1);
    }
    if (t >= 2) {
      const float d2 = (p - 2.0f * pp1 + pp2) - (g - 2.0f * tp1 + tp2);
      a_delta2 += fabsf(d2) * (m * m1 * m2);
    }
    if (d == 0) {               // mask counts (lane 0 only)
      c_m   += m;
      c_dm  += m * m1;
      c_d2m += m * m1 * m2;
    }

    // per-frame energy: sum_d (p - g) across 80 lanes (3 waves)
    float e = diff;
    #pragma unroll
    for (int o = 16; o >= 1; o >>= 1) e += __shfl_xor(e, o, 32);
    if ((tid & 31) == 0) ePart[i * 4 + (tid >> 5)] = e;

    pp2 = pp1; pp1 = p; tp2 = tp1; tp1 = g; m2 = m1; m1 = m;
  }
  __syncthreads();

  float a_energy = 0.f;
  if (tid < TT) {
    const int t = t0 + tid;
    if (t < TLEN) {
      const float s = ePart[tid * 4 + 0] + ePart[tid * 4 + 1] + ePart[tid * 4 + 2];
      a_energy = fabsf(s) * (1.0f / N_MELS) * mask[mbase + t];
    }
  }

  // wave-level exact f32 reductions via WMMA (EXEC all-ones here, no divergence)
  const float v0 = wave_sum32(a_l1);
  const float v1 = wave_sum32(a_band);
  const float v2 = wave_sum32(a_diff2);
  const float v3 = wave_sum32(a_tgt2);
  const float v4 = wave_sum32(a_delta);
  const float v5 = wave_sum32(a_delta2);
  const float v6 = wave_sum32(a_energy);

  const int w = tid >> 5;
  if ((tid & 31) == 0) {
    red[w * 8 + 0] = v0; red[w * 8 + 1] = v1; red[w * 8 + 2] = v2;
    red[w * 8 + 3] = v3; red[w * 8 + 4] = v4; red[w * 8 + 5] = v5;
    red[w * 8 + 6] = v6;
  }
  __syncthreads();
  if (tid == 0) {
    #pragma unroll
    for (int j = 0; j < 7; ++j)
      atomicAdd(&acc[j], red[0 * 8 + j] + red[1 * 8 + j] + red[2 * 8 + j]);
    atomicAdd(&acc[7], c_m);
    atomicAdd(&acc[8], c_dm);
    atomicAdd(&acc[9], c_d2m);
  }
}

__global__ void mel_loss_final(const float* __restrict__ bw,
                               const float* __restrict__ acc,
                               float* __restrict__ out) {
  if (threadIdx.x == 0) {
    float sw = 0.f;
    for (int i = 0; i < N_MELS; ++i) sw += bw[i];
    const float meanw = sw / (float)N_MELS;
    const float Nm = acc[7], Ndm = acc[8], Nd2 = acc[9];
    const float n1 = fmaxf(Nm * (float)N_MELS, 1.0f);
    const float l1     = acc[0] / n1;
    const float band   = (acc[1] / n1) / meanw;
    const float sc     = sqrtf(acc[2] / n1) / fmaxf(sqrtf(acc[3] / n1), 1e-8f);
    const float delta  = acc[4] / fmaxf(Ndm * (float)N_MELS, 1.0f);
    const float delta2 = acc[5] / fmaxf(Nd2 * (float)N_MELS, 1.0f);
    const float energy = acc[6] / fmaxf(Nm, 1.0f);
    out[0] = 1.0f * l1 + 0.5f * delta + 0.25f * delta2 +
             0.5f * sc + 1.0f * band + 0.5f * energy;
  }
}

extern "C" void kernel_launch(void* const* d_in, const int* in_sizes, int n_in,
                              void* d_out, int out_size, void* d_ws, size_t ws_size,
                              hipStream_t stream) {
  const float* pred = (const float*)d_in[0];
  const float* tgt  = (const float*)d_in[1];
  const float* mask = (const float*)d_in[2];
  const float* bw   = (const float*)d_in[3];
  float* acc = (float*)d_ws;
  float* out = (float*)d_out;

  mel_loss_init<<<dim3(1), dim3(32), 0, stream>>>(acc);
  dim3 grid(BATCH, (TLEN + TT - 1) / TT);
  mel_loss_main<<<grid, dim3(NTHREADS), 0, stream>>>(pred, tgt, mask, bw, acc);
  mel_loss_final<<<dim3(1), dim3(32), 0, stream>>>(bw, acc, out);
}